// Gemma3Main_33260226740348
// MI455X (gfx1250) — compile-verified
//
#include <hip/hip_runtime.h>
#include <hip/hip_bf16.h>
#include <cstddef>

typedef __attribute__((ext_vector_type(2))) float v2f;
typedef __attribute__((ext_vector_type(8))) float v8f;

#define NLAYERS 26
#define BQ 2
#define DM 1152
#define FF 4608
#define VV 32768
#define HDIM 256
#define SCACHE 2048
#define STOT 2049            // S + T
#define NHEADS 4
#define ATT_CHUNKS 16
#define TILES_TOTAL 129      // ceil(2049/16)
#define TILES_PER_CHUNK 8    // chunks 0..14 -> 8 tiles, chunk 15 -> 9 tiles

__device__ __forceinline__ int imin(int a, int b) { return a < b ? a : b; }

__device__ __forceinline__ float gelu_tanh(float v) {
    float v3 = v * v * v;
    return 0.5f * v * (1.0f + tanhf(0.7978845608f * (v + 0.044715f * v3)));
}

// ---------------------------------------------------------------- utilities
__global__ __launch_bounds__(256) void copy_kernel(float* __restrict__ dst,
                                                   const float* __restrict__ src, int n) {
    int i = blockIdx.x * 256 + threadIdx.x;
    if (i < n) dst[i] = src[i];
}

// out[b][j] = rms_norm(in[b], w)[j], rows of length C, one block per row
__global__ __launch_bounds__(256) void rmsnorm_kernel(float* __restrict__ out,
                                                      const float* __restrict__ in,
                                                      const float* __restrict__ w, int C) {
    const int b = blockIdx.x, tid = threadIdx.x;
    const float* row = in + (size_t)b * C;
    __shared__ float s_red[256];
    float ss = 0.f;
    for (int j = tid; j < C; j += 256) { float v = row[j]; ss += v * v; }
    s_red[tid] = ss; __syncthreads();
    for (int off = 128; off > 0; off >>= 1) {
        if (tid < off) s_red[tid] += s_red[tid + off];
        __syncthreads();
    }
    float inv = rsqrtf(s_red[0] / (float)C + 1e-6f);
    for (int j = tid; j < C; j += 256)
        out[(size_t)b * C + j] = row[j] * inv * (1.0f + w[j]);
}

// h[b][j] += rms_norm(y[b], w)[j]
__global__ __launch_bounds__(256) void addnorm_kernel(float* __restrict__ h,
                                                      const float* __restrict__ y,
                                                      const float* __restrict__ w, int C) {
    const int b = blockIdx.x, tid = threadIdx.x;
    const float* row = y + (size_t)b * C;
    __shared__ float s_red[256];
    float ss = 0.f;
    for (int j = tid; j < C; j += 256) { float v = row[j]; ss += v * v; }
    s_red[tid] = ss; __syncthreads();
    for (int off = 128; off > 0; off >>= 1) {
        if (tid < off) s_red[tid] += s_red[tid + off];
        __syncthreads();
    }
    float inv = rsqrtf(s_red[0] / (float)C + 1e-6f);
    for (int j = tid; j < C; j += 256)
        h[(size_t)b * C + j] += row[j] * inv * (1.0f + w[j]);
}

// --------------------------------------------------------- K-split GEMV
// part[(ks*2+b)*N + n] = sum over K-chunk ks of x[b][k]*W[k][n]
// W is (K,N) row-major; K assumed divisible by gridDim.y; chunk <= 1152.
// Weight stream is read exactly once per call -> non-temporal loads.
__global__ __launch_bounds__(256) void gemv_part_kernel(float* __restrict__ part,
                                                        const float* __restrict__ x,
                                                        const float* __restrict__ W,
                                                        int K, int N) {
    __shared__ float sx0[1152];
    __shared__ float sx1[1152];
    const int nks = gridDim.y;
    const int ks = blockIdx.y;
    const int kc = K / nks;
    const int k0 = ks * kc;
    const int tid = threadIdx.x;
    for (int j = tid; j < kc; j += 256) {
        sx0[j] = x[k0 + j];
        sx1[j] = x[K + k0 + j];
    }
    __syncthreads();
    const int n = blockIdx.x * 256 + tid;
    if (n >= N) return;
    float a0 = 0.f, a1 = 0.f;
    const float* wp = W + (size_t)k0 * N + n;
    for (int k = 0; k < kc; ++k) {
        float wv = __builtin_nontemporal_load(wp + (size_t)k * N);
        a0 += sx0[k] * wv;
        a1 += sx1[k] * wv;
    }
    part[((size_t)ks * 2 + 0) * N + n] = a0;
    part[((size_t)ks * 2 + 1) * N + n] = a1;
}

// y[b][n] = sum_ks part[(ks*2+b)*N+n]   (deterministic fixed-order reduce)
__global__ __launch_bounds__(256) void combine_kernel(float* __restrict__ y,
                                                      const float* __restrict__ part,
                                                      int N, int nks) {
    int idx = blockIdx.x * 256 + threadIdx.x;
    if (idx >= 2 * N) return;
    int b = idx / N, n = idx - b * N;
    float s = 0.f;
    for (int ks = 0; ks < nks; ++ks) s += part[((size_t)ks * 2 + b) * N + n];
    y[idx] = s;
}

// ff[b][n] = gelu_tanh(sum gate partials) * (sum up partials)
__global__ __launch_bounds__(256) void geluup_kernel(float* __restrict__ ff,
                                                     const float* __restrict__ pg,
                                                     const float* __restrict__ pu,
                                                     int N, int nks) {
    int idx = blockIdx.x * 256 + threadIdx.x;
    if (idx >= 2 * N) return;
    int b = idx / N, n = idx - b * N;
    float g = 0.f, u = 0.f;
    for (int ks = 0; ks < nks; ++ks) {
        g += pg[((size_t)ks * 2 + b) * N + n];
        u += pu[((size_t)ks * 2 + b) * N + n];
    }
    ff[idx] = gelu_tanh(g) * u;
}

// ------------------------------------------------- QK RMSNorm + RoPE
// grid = 10: b = blk/5, r = blk%5 (r<4: q head r, r==4: k). 256 threads = HD.
__global__ __launch_bounds__(256) void qknorm_rope_kernel(const float* __restrict__ qkv,
                                                          const float* __restrict__ qw,
                                                          const float* __restrict__ kw,
                                                          const float* __restrict__ cosv,
                                                          const float* __restrict__ sinv,
                                                          float* __restrict__ qn,
                                                          float* __restrict__ kn) {
    const int blk = blockIdx.x;
    const int b = blk / 5, r = blk % 5;
    const int d = threadIdx.x;
    const bool isK = (r == 4);
    const float* src = qkv + b * 1536 + (isK ? 1024 : r * HDIM);
    const float* w = isK ? kw : qw;
    __shared__ float s_x[HDIM];
    __shared__ float s_red[HDIM];
    float xv = src[d];
    s_red[d] = xv * xv;
    __syncthreads();
    for (int off = 128; off > 0; off >>= 1) {
        if (d < off) s_red[d] += s_red[d + off];
        __syncthreads();
    }
    float inv = rsqrtf(s_red[0] / (float)HDIM + 1e-6f);
    float xn = xv * inv * (1.0f + w[d]);
    s_x[d] = xn;
    __syncthreads();
    float rot = (d < 128) ? -s_x[d + 128] : s_x[d - 128];
    float o = xn * cosv[d] + rot * sinv[d];
    if (isK) kn[b * HDIM + d] = o;
    else     qn[(b * NHEADS + r) * HDIM + d] = o;
}

// ------------------------------------------------- attention (flash-decode)
// grid (ATT_CHUNKS, B), 256 threads (8 waves). WMMA f32 16x16x4 for scores:
// A = q padded to 16 rows (heads 0-3), B = K-cache tile (4 k-values x 16 pos).
// Padding rows (srow > 2048) are clamped to the (valid, global) new-k row:
// their score columns are never read, and keeping krow strictly global lets
// the compiler emit global_load_b64 (not flat_load) in the WMMA hot loop.
__global__ __launch_bounds__(256) void attn_part_kernel(const float* __restrict__ qn,
                                                        const float* __restrict__ knew,
                                                        const float* __restrict__ qkv,
                                                        const float* __restrict__ kcache,
                                                        const float* __restrict__ vcache,
                                                        const float* __restrict__ mask,
                                                        float* __restrict__ pm,
                                                        float* __restrict__ pl,
                                                        float* __restrict__ pacc) {
    const int c = blockIdx.x;
    const int b = blockIdx.y;
    const int t0 = c * TILES_PER_CHUNK;
    const int t1 = (c == ATT_CHUNKS - 1) ? TILES_TOTAL : (t0 + TILES_PER_CHUNK);
    const int base = t0 * 16;
    const int npos = (t1 - t0) * 16;              // 128 or 144
    const int valid = imin(STOT - base, npos);

    const int tid = threadIdx.x;
    const int lane = tid & 31;
    const int wave = tid >> 5;

    __shared__ float s_q[16][HDIM];   // rows 4..15 zero
    __shared__ float s_sc[NHEADS][144];
    __shared__ float s_red[256];

    for (int i = tid; i < 16 * HDIM; i += 256) {
        int r = i >> 8, d = i & 255;
        s_q[r][d] = (r < NHEADS) ? qn[(b * NHEADS + r) * HDIM + d] : 0.0f;
    }
    __syncthreads();

    // ---- phase 1: scores via V_WMMA_F32_16X16X4_F32 ----
    const int half = lane >> 4;       // 0: lanes 0-15, 1: lanes 16-31
    const int lr = lane & 15;         // A row M / B col N for this lane
    const int koff = half * 2;        // K sub-offset held by this half-wave
    const float* arow = &s_q[lr][0];

    for (int t = t0 + wave; t < t1; t += 8) {
        const int srow = t * 16 + lr;
        // srow >= 2048 (new token AND dead padding columns) -> new-k row.
        const float* krow = (srow < SCACHE)
                                ? (kcache + ((size_t)b * SCACHE + srow) * HDIM)
                                : (knew + b * HDIM);
        v8f cacc = {};
#pragma unroll 4
        for (int k = 0; k < HDIM; k += 4) {
            v2f av, bv;
            av.x = arow[k + koff];
            av.y = arow[k + koff + 1];
            bv.x = __builtin_nontemporal_load(krow + k + koff);
            bv.y = __builtin_nontemporal_load(krow + k + koff + 1);
            cacc = __builtin_amdgcn_wmma_f32_16x16x4_f32(
                false, av, false, bv, (short)0, cacc, false, false);
        }
        if (half == 0) {            // C: VGPR r = row M=r for lanes 0-15
#pragma unroll
            for (int r = 0; r < NHEADS; ++r)
                s_sc[r][(t - t0) * 16 + lr] = cacc[r] * 0.0625f;  // * 1/sqrt(HD)
        }
    }
    __syncthreads();

    // ---- phase 2: softcap + mask + local softmax stats ----
    float mh[NHEADS], lh[NHEADS];
    for (int hh = 0; hh < NHEADS; ++hh) {
        const bool has = (tid < valid);
        float sv = 0.0f, lm = -3.0e38f;
        if (has) {
            sv = s_sc[hh][tid];
            sv = tanhf(sv * 0.02f) * 50.0f + mask[b * STOT + base + tid];
            lm = sv;
        }
        s_red[tid] = lm; __syncthreads();
        for (int off = 128; off > 0; off >>= 1) {
            if (tid < off) s_red[tid] = fmaxf(s_red[tid], s_red[tid + off]);
            __syncthreads();
        }
        float m = s_red[0]; __syncthreads();
        float e = has ? expf(sv - m) : 0.0f;
        if (tid < npos) s_sc[hh][tid] = e;   // zero in padding region
        s_red[tid] = e; __syncthreads();
        for (int off = 128; off > 0; off >>= 1) {
            if (tid < off) s_red[tid] += s_red[tid + off];
            __syncthreads();
        }
        mh[hh] = m; lh[hh] = s_red[0];
        __syncthreads();
    }

    // ---- phase 3: partial attn output, thread = d, V rows contiguous in s ----
    const int d = tid;
    float acc[NHEADS] = {0.f, 0.f, 0.f, 0.f};
    const float* vrow = vcache + ((size_t)b * HDIM + d) * SCACHE;
    const int cache_end = imin(valid, SCACHE - base);
    for (int j = 0; j < cache_end; ++j) {
        float vv = __builtin_nontemporal_load(vrow + base + j);
#pragma unroll
        for (int hh = 0; hh < NHEADS; ++hh) acc[hh] += s_sc[hh][j] * vv;
    }
    if (base + valid == STOT) {              // chunk contains the new token
        float vv = qkv[b * 1536 + 1280 + d]; // v_new
        int j = SCACHE - base;
#pragma unroll
        for (int hh = 0; hh < NHEADS; ++hh) acc[hh] += s_sc[hh][j] * vv;
    }
    const int cb = b * ATT_CHUNKS + c;
#pragma unroll
    for (int hh = 0; hh < NHEADS; ++hh)
        pacc[((size_t)cb * NHEADS + hh) * HDIM + d] = acc[hh];
    if (tid < NHEADS) { pm[cb * NHEADS + tid] = mh[tid]; pl[cb * NHEADS + tid] = lh[tid]; }
}

// merge chunk partials (fixed order -> deterministic)
__global__ __launch_bounds__(256) void attn_combine_kernel(const float* __restrict__ pm,
                                                           const float* __restrict__ pl,
                                                           const float* __restrict__ pacc,
                                                           float* __restrict__ attn_out) {
    const int b = blockIdx.x;
    const int d = threadIdx.x;
    for (int hh = 0; hh < NHEADS; ++hh) {
        float M = -3.0e38f;
        for (int cc = 0; cc < ATT_CHUNKS; ++cc)
            M = fmaxf(M, pm[(b * ATT_CHUNKS + cc) * NHEADS + hh]);
        float L = 0.f, A = 0.f;
        for (int cc = 0; cc < ATT_CHUNKS; ++cc) {
            const int cb = b * ATT_CHUNKS + cc;
            float w = expf(pm[cb * NHEADS + hh] - M);
            L += pl[cb * NHEADS + hh] * w;
            A += pacc[((size_t)cb * NHEADS + hh) * HDIM + d] * w;
        }
        attn_out[b * (NHEADS * HDIM) + hh * HDIM + d] = A / L;
    }
}

// =====================================================================
extern "C" void kernel_launch(void* const* d_in, const int* in_sizes, int n_in,
                              void* d_out, int out_size, void* d_ws, size_t ws_size,
                              hipStream_t stream) {
    (void)in_sizes; (void)n_in; (void)out_size; (void)ws_size;

    const float* embeddings = (const float*)d_in[0];
    const float* mask_g     = (const float*)d_in[1];
    const float* mask_l     = (const float*)d_in[2];
    const float* cos_g      = (const float*)d_in[3];
    const float* sin_g      = (const float*)d_in[4];
    const float* cos_l      = (const float*)d_in[5];
    const float* sin_l      = (const float*)d_in[6];
    const float* kcache     = (const float*)d_in[7];
    const float* vcache     = (const float*)d_in[8];
    const float* pre_attn_w = (const float*)d_in[9];
    const float* qkv_w      = (const float*)d_in[10];
    const float* qnw        = (const float*)d_in[11];
    const float* knw        = (const float*)d_in[12];
    const float* out_w      = (const float*)d_in[13];
    const float* post_attn_w= (const float*)d_in[14];
    const float* pre_ff_w   = (const float*)d_in[15];
    const float* gate_w     = (const float*)d_in[16];
    const float* up_w       = (const float*)d_in[17];
    const float* down_w     = (const float*)d_in[18];
    const float* post_ff_w  = (const float*)d_in[19];
    const float* final_w    = (const float*)d_in[20];
    const float* lm_w       = (const float*)d_in[21];

    float* ws   = (float*)d_ws;
    float* h    = ws; ws += BQ * DM;            // 2304
    float* xb   = ws; ws += BQ * DM;            // 2304
    float* qkv  = ws; ws += BQ * 1536;          // 3072
    float* qn   = ws; ws += BQ * NHEADS * HDIM; // 2048
    float* kn   = ws; ws += BQ * HDIM;          // 512
    float* attn = ws; ws += BQ * NHEADS * HDIM; // 2048
    float* yb   = ws; ws += BQ * DM;            // 2304
    float* ffb  = ws; ws += BQ * FF;            // 9216
    float* pm   = ws; ws += BQ * ATT_CHUNKS * NHEADS;          // 128
    float* pl   = ws; ws += BQ * ATT_CHUNKS * NHEADS;          // 128
    float* pacc = ws; ws += BQ * ATT_CHUNKS * NHEADS * HDIM;   // 32768
    float* partA= ws; ws += 4 * 2 * VV;         // 262144 (covers all uses)
    float* partB= ws; ws += 8 * 2 * FF;         // 73728

    copy_kernel<<<(BQ * DM + 255) / 256, 256, 0, stream>>>(h, embeddings, BQ * DM);

    for (int i = 0; i < NLAYERS; ++i) {
        const bool loc = ((i + 1) % 6) == 0;
        const float* cosv = loc ? cos_l : cos_g;
        const float* sinv = loc ? sin_l : sin_g;
        const float* mask = loc ? mask_l : mask_g;

        // pre-attn norm
        rmsnorm_kernel<<<BQ, 256, 0, stream>>>(xb, h, pre_attn_w + i * DM, DM);

        // qkv projection: (2,1152) @ (1152,1536)
        gemv_part_kernel<<<dim3(6, 8), 256, 0, stream>>>(
            partA, xb, qkv_w + (size_t)i * DM * 1536, DM, 1536);
        combine_kernel<<<12, 256, 0, stream>>>(qkv, partA, 1536, 8);

        // q/k rmsnorm + rope
        qknorm_rope_kernel<<<10, 256, 0, stream>>>(
            qkv, qnw + i * HDIM, knw + i * HDIM, cosv, sinv, qn, kn);

        // attention (flash-decode over 16 S-chunks per batch)
        attn_part_kernel<<<dim3(ATT_CHUNKS, BQ), 256, 0, stream>>>(
            qn, kn, qkv,
            kcache + (size_t)i * BQ * SCACHE * HDIM,
            vcache + (size_t)i * BQ * HDIM * SCACHE,
            mask, pm, pl, pacc);
        attn_combine_kernel<<<BQ, 256, 0, stream>>>(pm, pl, pacc, attn);

        // output projection: (2,1024) @ (1024,1152)
        gemv_part_kernel<<<dim3(5, 8), 256, 0, stream>>>(
            partA, attn, out_w + (size_t)i * (NHEADS * HDIM) * DM, NHEADS * HDIM, DM);
        combine_kernel<<<9, 256, 0, stream>>>(yb, partA, DM, 8);

        addnorm_kernel<<<BQ, 256, 0, stream>>>(h, yb, post_attn_w + i * DM, DM);

        // MLP
        rmsnorm_kernel<<<BQ, 256, 0, stream>>>(xb, h, pre_ff_w + i * DM, DM);
        gemv_part_kernel<<<dim3(18, 8), 256, 0, stream>>>(
            partA, xb, gate_w + (size_t)i * DM * FF, DM, FF);
        gemv_part_kernel<<<dim3(18, 8), 256, 0, stream>>>(
            partB, xb, up_w + (size_t)i * DM * FF, DM, FF);
        geluup_kernel<<<36, 256, 0, stream>>>(ffb, partA, partB, FF, 8);
        gemv_part_kernel<<<dim3(5, 8), 256, 0, stream>>>(
            partA, ffb, down_w + (size_t)i * FF * DM, FF, DM);
        combine_kernel<<<9, 256, 0, stream>>>(yb, partA, DM, 8);

        addnorm_kernel<<<BQ, 256, 0, stream>>>(h, yb, post_ff_w + i * DM, DM);
    }

    // final norm + lm head
    rmsnorm_kernel<<<BQ, 256, 0, stream>>>(xb, h, final_w, DM);
    gemv_part_kernel<<<dim3(VV / 256, 4), 256, 0, stream>>>(partA, xb, lm_w, DM, VV);
    combine_kernel<<<(2 * VV + 255) / 256, 256, 0, stream>>>((float*)d_out, partA, VV, 4);
}